// BoxOnlyHungarianMatcher_47588237639953
// MI455X (gfx1250) — compile-verified
//
#include <hip/hip_runtime.h>
#include <hip/hip_bf16.h>

// ---------------------------------------------------------------------------
// BoxOnlyHungarianMatcher cost matrix for MI455X (gfx1250).
//   cost[r, m] = 5 * L1(pred_cxcywh[r], gt_cxcywh[m]) - 2 * GIoU(corners)
// N = 14400 pred rows, M = 1600 gt cols, output 92.2 MB f32 -> store-BW bound
// (~4 us floor at 23.3 TB/s). No bilinear structure -> WMMA inapplicable;
// the CDNA5-specific path exercised is the async data mover:
// global_load_async_to_lds_b128 (+ s_wait_asynccnt) staging GT boxes into
// LDS, SoA layout for conflict-free ds_load_b128, and non-temporal b128
// stores for the write-once output stream.
// ---------------------------------------------------------------------------

typedef float vf4 __attribute__((ext_vector_type(4)));
typedef int   vi4 __attribute__((ext_vector_type(4)));

#define AS1 __attribute__((address_space(1)))
#define AS3 __attribute__((address_space(3)))

#ifndef __has_builtin
#define __has_builtin(x) 0
#endif

#if __has_builtin(__builtin_amdgcn_global_load_async_to_lds_b128)
#define USE_ASYNC_LDS 1
#else
#define USE_ASYNC_LDS 0
#endif

static constexpr int MAX_M    = 1600;              // GT capacity held in LDS
static constexpr int BLOCK    = 256;               // 8 waves (wave32)
static constexpr int WAVE     = 32;
static constexpr int NWAVES   = BLOCK / WAVE;      // 8
static constexpr int ROWS_PW  = 4;                 // rows handled per wave
static constexpr int ROWS_PB  = NWAVES * ROWS_PW;  // 32 rows per block
static constexpr int REGCHUNK = (MAX_M + BLOCK - 1) / BLOCK; // 7

// LDS: stage raw AoS cxcywh, then overlay with derived SoA (barrier between).
union GtSmem {
    vf4 staging[MAX_M];                       // 25.6 KB raw cxcywh
    struct {
        float cx[MAX_M], cy[MAX_M], w[MAX_M], h[MAX_M];
        float x0[MAX_M], y0[MAX_M], x1[MAX_M], y1[MAX_M];
        float area[MAX_M];
    } soa;                                    // 57.6 KB
};

__device__ __forceinline__ float pair_cost(
    float pcx, float pcy, float pw, float ph,
    float px0, float py0, float px1, float py1, float parea,
    float gcx, float gcy, float gw, float gh,
    float gx0, float gy0, float gx1, float gy1, float garea)
{
    // L1 on cxcywh (abs-modifier friendly: subs + adds with |src|)
    float l1 = fabsf(pcx - gcx) + fabsf(pcy - gcy) +
               fabsf(pw  - gw ) + fabsf(ph  - gh );

    // Intersection
    float iw = fminf(px1, gx1) - fmaxf(px0, gx0);
    float ih = fminf(py1, gy1) - fmaxf(py0, gy0);
    iw = fmaxf(iw, 0.0f);
    ih = fmaxf(ih, 0.0f);
    float inter = iw * ih;
    float uni   = parea + garea - inter;
    float iou   = inter * __builtin_amdgcn_rcpf(uni);

    // Enclosing box
    float ew = fmaxf(px1, gx1) - fminf(px0, gx0);
    float eh = fmaxf(py1, gy1) - fminf(py0, gy0);
    ew = fmaxf(ew, 0.0f);
    eh = fmaxf(eh, 0.0f);
    float earea = ew * eh;
    float giou  = iou - (earea - uni) * __builtin_amdgcn_rcpf(earea);

    return 5.0f * l1 - 2.0f * giou;
}

__global__ __launch_bounds__(BLOCK)
void matcher_cost_kernel(const float* __restrict__ pred,
                         const float* __restrict__ gt,
                         float* __restrict__ out,
                         int N, int M)
{
    __shared__ GtSmem sm;
    const int tid  = threadIdx.x;
    const int lane = tid & (WAVE - 1);
    const int wid  = tid / WAVE;

    // ---- Phase 1: async-DMA GT boxes (AoS cxcywh) into LDS ----
    const vf4* gt4 = (const vf4*)gt;
    for (int i = tid; i < M; i += BLOCK) {
#if USE_ASYNC_LDS
        __builtin_amdgcn_global_load_async_to_lds_b128(
            (AS1 vi4*)(gt4 + i),
            (AS3 vi4*)&sm.staging[i],
            /*offset=*/0, /*cpol=*/0);
#else
        sm.staging[i] = gt4[i];
#endif
    }
#if USE_ASYNC_LDS
#if __has_builtin(__builtin_amdgcn_s_wait_asynccnt)
    __builtin_amdgcn_s_wait_asynccnt(0);
#else
    asm volatile("s_wait_asynccnt 0" ::: "memory");
#endif
    asm volatile("" ::: "memory"); // keep ds reads below the wait
#endif

    // Each thread reads back exactly the elements it issued (per-wave
    // asynccnt wait is sufficient), then the whole block transposes to SoA
    // over the same LDS bytes (barrier protects the overlay).
    vf4 rv[REGCHUNK];
#pragma unroll
    for (int k = 0; k < REGCHUNK; ++k) {
        int i = tid + k * BLOCK;
        if (i < M) rv[k] = sm.staging[i];
    }
    __syncthreads();
#pragma unroll
    for (int k = 0; k < REGCHUNK; ++k) {
        int i = tid + k * BLOCK;
        if (i < M) {
            vf4 b = rv[k];
            float x0 = b.x - 0.5f * b.z, y0 = b.y - 0.5f * b.w;
            float x1 = b.x + 0.5f * b.z, y1 = b.y + 0.5f * b.w;
            sm.soa.cx[i] = b.x; sm.soa.cy[i] = b.y;
            sm.soa.w[i]  = b.z; sm.soa.h[i]  = b.w;
            sm.soa.x0[i] = x0;  sm.soa.y0[i] = y0;
            sm.soa.x1[i] = x1;  sm.soa.y1[i] = y1;
            sm.soa.area[i] = (x1 - x0) * (y1 - y0);
        }
    }
    __syncthreads();

    // ---- Phase 2: one wave per output row, 4 gt columns per lane ----
    const int rowBase = blockIdx.x * ROWS_PB + wid * ROWS_PW;
    for (int rr = 0; rr < ROWS_PW; ++rr) {
        const int r = rowBase + rr;        // wave-uniform
        if (r >= N) break;

        vf4 pb = ((const vf4*)pred)[r];    // wave-uniform load
        const float pcx = pb.x, pcy = pb.y, pw = pb.z, ph = pb.w;
        const float px0 = pcx - 0.5f * pw, py0 = pcy - 0.5f * ph;
        const float px1 = pcx + 0.5f * pw, py1 = pcy + 0.5f * ph;
        const float parea = (px1 - px0) * (py1 - py0);

        float* __restrict__ orow = out + (size_t)r * (size_t)M;

        for (int base = 0; base < M; base += WAVE * 4) {
            const int m0 = base + lane * 4;
            if (m0 + 3 < M) {
                // Contiguous 512B per SoA array across the wave: conflict-free
                vf4 vcx = *(const vf4*)&sm.soa.cx[m0];
                vf4 vcy = *(const vf4*)&sm.soa.cy[m0];
                vf4 vw  = *(const vf4*)&sm.soa.w[m0];
                vf4 vh  = *(const vf4*)&sm.soa.h[m0];
                vf4 vx0 = *(const vf4*)&sm.soa.x0[m0];
                vf4 vy0 = *(const vf4*)&sm.soa.y0[m0];
                vf4 vx1 = *(const vf4*)&sm.soa.x1[m0];
                vf4 vy1 = *(const vf4*)&sm.soa.y1[m0];
                vf4 va  = *(const vf4*)&sm.soa.area[m0];
                vf4 res;
                res.x = pair_cost(pcx, pcy, pw, ph, px0, py0, px1, py1, parea,
                                  vcx.x, vcy.x, vw.x, vh.x, vx0.x, vy0.x, vx1.x, vy1.x, va.x);
                res.y = pair_cost(pcx, pcy, pw, ph, px0, py0, px1, py1, parea,
                                  vcx.y, vcy.y, vw.y, vh.y, vx0.y, vy0.y, vx1.y, vy1.y, va.y);
                res.z = pair_cost(pcx, pcy, pw, ph, px0, py0, px1, py1, parea,
                                  vcx.z, vcy.z, vw.z, vh.z, vx0.z, vy0.z, vx1.z, vy1.z, va.z);
                res.w = pair_cost(pcx, pcy, pw, ph, px0, py0, px1, py1, parea,
                                  vcx.w, vcy.w, vw.w, vh.w, vx0.w, vy0.w, vx1.w, vy1.w, va.w);
                // Write-once 92MB stream: non-temporal b128 store
                __builtin_nontemporal_store(res, (vf4*)(orow + m0));
            } else if (m0 < M) {
                for (int j = 0; j < 4; ++j) {
                    int m = m0 + j;
                    if (m < M) {
                        float c = pair_cost(pcx, pcy, pw, ph, px0, py0, px1, py1, parea,
                                            sm.soa.cx[m], sm.soa.cy[m], sm.soa.w[m], sm.soa.h[m],
                                            sm.soa.x0[m], sm.soa.y0[m], sm.soa.x1[m], sm.soa.y1[m],
                                            sm.soa.area[m]);
                        __builtin_nontemporal_store(c, orow + m);
                    }
                }
            }
        }
    }
}

// Fallback for M > MAX_M (not hit by the reference shapes): no LDS staging.
__global__ void matcher_cost_naive(const float* __restrict__ pred,
                                   const float* __restrict__ gt,
                                   float* __restrict__ out,
                                   int N, int M)
{
    long long idx   = (long long)blockIdx.x * blockDim.x + threadIdx.x;
    long long total = (long long)N * (long long)M;
    if (idx >= total) return;
    int r = (int)(idx / M);
    int m = (int)(idx % M);

    vf4 pb = ((const vf4*)pred)[r];
    vf4 gb = ((const vf4*)gt)[m];
    float px0 = pb.x - 0.5f * pb.z, py0 = pb.y - 0.5f * pb.w;
    float px1 = pb.x + 0.5f * pb.z, py1 = pb.y + 0.5f * pb.w;
    float gx0 = gb.x - 0.5f * gb.z, gy0 = gb.y - 0.5f * gb.w;
    float gx1 = gb.x + 0.5f * gb.z, gy1 = gb.y + 0.5f * gb.w;
    float parea = (px1 - px0) * (py1 - py0);
    float garea = (gx1 - gx0) * (gy1 - gy0);
    float c = pair_cost(pb.x, pb.y, pb.z, pb.w, px0, py0, px1, py1, parea,
                        gb.x, gb.y, gb.z, gb.w, gx0, gy0, gx1, gy1, garea);
    __builtin_nontemporal_store(c, out + idx);
}

extern "C" void kernel_launch(void* const* d_in, const int* in_sizes, int n_in,
                              void* d_out, int out_size, void* d_ws, size_t ws_size,
                              hipStream_t stream) {
    const float* pred = (const float*)d_in[0];   // [16,900,4] f32 cxcywh
    const float* gt   = (const float*)d_in[1];   // [1600,4]  f32 cxcywh
    float* out = (float*)d_out;                  // [16,900,1600] f32

    const int N = in_sizes[0] / 4;               // 14400 pred rows
    const int M = in_sizes[1] / 4;               // 1600 gt cols

    if (M <= MAX_M) {
        dim3 grid((N + ROWS_PB - 1) / ROWS_PB);  // 450 blocks
        matcher_cost_kernel<<<grid, BLOCK, 0, stream>>>(pred, gt, out, N, M);
    } else {
        long long total = (long long)N * (long long)M;
        dim3 grid((unsigned)((total + BLOCK - 1) / BLOCK));
        matcher_cost_naive<<<grid, BLOCK, 0, stream>>>(pred, gt, out, N, M);
    }
}